// MambaBlock_14267881357438
// MI455X (gfx1250) — compile-verified
//
#include <hip/hip_runtime.h>
#include <hip/hip_bf16.h>
#include <math.h>

// ---------------- problem constants ----------------
#define BATCH   2
#define SEQ     2048
#define DMODEL  1024
#define INNER   2048
#define DSTATE  16
#define DTRANK  64
#define MROWS   (BATCH * SEQ)      // 4096 rows for all GEMMs

// ---------------- workspace layout (floats) ----------------
#define XN_OFF   ((size_t)0)                               // 4096*1024
#define UZ_OFF   (XN_OFF  + (size_t)MROWS * DMODEL)        // 4096*4096
#define U_OFF    (UZ_OFF  + (size_t)MROWS * 2 * INNER)     // 4096*2048 (also y*silu(z))
#define DTR_OFF  (U_OFF   + (size_t)MROWS * INNER)         // 4096*64
#define DT_OFF   (DTR_OFF + (size_t)MROWS * DTRANK)        // 4096*2048
#define BM_OFF   (DT_OFF  + (size_t)MROWS * INNER)         // 4096*16
#define CM_OFF   (BM_OFF  + (size_t)MROWS * DSTATE)        // 4096*16

typedef __attribute__((ext_vector_type(2))) float v2f;
typedef __attribute__((ext_vector_type(8))) float v8f;

__device__ __forceinline__ float sigmoidf_(float x) { return 1.0f / (1.0f + __expf(-x)); }
__device__ __forceinline__ float siluf_(float x)    { return x * sigmoidf_(x); }

// Async global->LDS copies (CDNA5 path, tracked by ASYNCcnt).
// LDS byte address = low 32 bits of the flat shared pointer (aperture rule).
__device__ __forceinline__ void async_copy_b128(float* lds_dst, const float* gsrc) {
    asm volatile("global_load_async_to_lds_b128 %0, %1, off"
                 :: "v"((unsigned int)(unsigned long long)lds_dst),
                    "v"((unsigned long long)gsrc)
                 : "memory");
}
__device__ __forceinline__ void async_copy_b32(float* lds_dst, const float* gsrc) {
    asm volatile("global_load_async_to_lds_b32 %0, %1, off"
                 :: "v"((unsigned int)(unsigned long long)lds_dst),
                    "v"((unsigned long long)gsrc)
                 : "memory");
}
__device__ __forceinline__ void async_wait_all() {
    asm volatile("s_wait_asynccnt 0x0" ::: "memory");
}

// ================= RMSNorm =================
__global__ void rmsnorm_kernel(const float* __restrict__ x,
                               const float* __restrict__ scale,
                               float* __restrict__ xn) {
    const int row = blockIdx.x;          // 0..4095
    const int tid = threadIdx.x;         // 256 threads
    __shared__ float red[256];
    float s = 0.f;
    for (int i = tid; i < DMODEL; i += 256) {
        float v = x[(size_t)row * DMODEL + i];
        s += v * v;
    }
    red[tid] = s;
    __syncthreads();
    for (int off = 128; off > 0; off >>= 1) {
        if (tid < off) red[tid] += red[tid + off];
        __syncthreads();
    }
    const float inv = rsqrtf(red[0] * (1.0f / DMODEL) + 1e-6f);
    for (int i = tid; i < DMODEL; i += 256) {
        xn[(size_t)row * DMODEL + i] = x[(size_t)row * DMODEL + i] * inv * scale[i];
    }
}

// ================= WMMA fp32 GEMM =================
// C[M,N] = act( A[M,K] @ B[K,N] + bias0 + bias1 ) + resid
// Block: 128 threads = 4 waves; block tile 64(M) x 16*NT(N); K-step 32.
// Double-buffered async global->LDS staging; B is transposed in-flight by the
// async engine (per-lane LDS destinations) so fragments are single b64 loads.
// ACT: 0 = none, 1 = tanh, 2 = softplus+clip(DT_MIN,DT_MAX)
template <int NT, int ACT>
__global__ __launch_bounds__(128)
void gemm_wmma_f32(const float* __restrict__ A, const float* __restrict__ Bm,
                   const float* __restrict__ bias0, const float* __restrict__ bias1,
                   const float* __restrict__ resid, float* __restrict__ C,
                   int M, int N, int K) {
    constexpr int BN = 16 * NT;
    constexpr int KSTEP = 32;
    constexpr int BTS = KSTEP + 2;       // Bt row stride: keeps 8B align, spreads banks
    __shared__ float As[2][64 * KSTEP];
    __shared__ float Bt[2][BN * BTS];

    const int tid  = threadIdx.x;
    const int lane = tid & 31;
    const int wave = tid >> 5;          // 0..3 -> 16-row slice
    const int m0 = blockIdx.y * 64;
    const int n0 = blockIdx.x * BN;

    const int mrow  = lane & 15;        // M index within 16 (also N index for B/C/D)
    const int khalf = lane >> 4;        // selects K pair (0 -> K{0,1}, 1 -> K{2,3})

    v8f acc[NT];
    #pragma unroll
    for (int t = 0; t < NT; ++t) acc[t] = (v8f){0,0,0,0,0,0,0,0};

    // ---- software pipeline: prime buffer 0 with the first K-tile
    {
        const int kt = 0;
        #pragma unroll 2
        for (int i4 = tid; i4 < (64 * KSTEP) / 4; i4 += 128) {
            const int r = i4 / (KSTEP / 4), c = (i4 % (KSTEP / 4)) * 4;
            async_copy_b128(&As[0][r * KSTEP + c],
                            &A[(size_t)(m0 + r) * K + (kt + c)]);
        }
        for (int e = tid; e < KSTEP * BN; e += 128) {
            const int k = e / BN, n = e % BN;   // consecutive lanes -> consecutive n (coalesced)
            async_copy_b32(&Bt[0][n * BTS + k],
                           &Bm[(size_t)(kt + k) * N + (n0 + n)]);
        }
    }

    int buf = 0;
    for (int k0 = 0; k0 < K; k0 += KSTEP) {
        async_wait_all();      // our tile (issued last iteration) has landed
        __syncthreads();       // everyone's tile has landed; prev readers done

        // issue next tile into the other buffer while we compute this one
        if (k0 + KSTEP < K) {
            const int kt = k0 + KSTEP, nb = buf ^ 1;
            #pragma unroll 2
            for (int i4 = tid; i4 < (64 * KSTEP) / 4; i4 += 128) {
                const int r = i4 / (KSTEP / 4), c = (i4 % (KSTEP / 4)) * 4;
                async_copy_b128(&As[nb][r * KSTEP + c],
                                &A[(size_t)(m0 + r) * K + (kt + c)]);
            }
            for (int e = tid; e < KSTEP * BN; e += 128) {
                const int k = e / BN, n = e % BN;
                async_copy_b32(&Bt[nb][n * BTS + k],
                               &Bm[(size_t)(kt + k) * N + (n0 + n)]);
            }
        }

        const float* __restrict__ Ab = &As[buf][0];
        const float* __restrict__ Bb = &Bt[buf][0];
        #pragma unroll
        for (int kk = 0; kk < KSTEP; kk += 4) {
            const int ke = kk + 2 * khalf;
            v2f a = *(const v2f*)(Ab + (wave * 16 + mrow) * KSTEP + ke);
            #pragma unroll
            for (int nt = 0; nt < NT; ++nt) {
                v2f b = *(const v2f*)(Bb + (nt * 16 + mrow) * BTS + ke);
                acc[nt] = __builtin_amdgcn_wmma_f32_16x16x4_f32(
                    false, a, false, b, (short)0, acc[nt], false, false);
            }
        }
        buf ^= 1;
    }

    // epilogue: D layout -> VGPR r holds row (r + 8*(lane/16)), col = lane%16
    #pragma unroll
    for (int nt = 0; nt < NT; ++nt) {
        const int col = n0 + nt * 16 + mrow;
        #pragma unroll
        for (int r = 0; r < 8; ++r) {
            const int row = m0 + wave * 16 + r + 8 * khalf;
            float v = acc[nt][r];
            if (bias0) v += bias0[col];
            if (bias1) v += bias1[col];
            if constexpr (ACT == 1) {
                v = tanhf(v);
            } else if constexpr (ACT == 2) {
                float sp = (v > 20.0f) ? v : log1pf(__expf(v));
                v = fminf(fmaxf(sp, 0.001f), 0.1f);
            }
            if (resid) v += resid[(size_t)row * N + col];
            C[(size_t)row * N + col] = v;
        }
    }
}

// ================= causal depthwise conv(4) + SiLU =================
__global__ void conv_silu_kernel(const float* __restrict__ uz,
                                 const float* __restrict__ ck,
                                 const float* __restrict__ cb,
                                 float* __restrict__ u) {
    const int idx = blockIdx.x * 256 + threadIdx.x;   // (b*SEQ + t)*INNER + d
    const int d  = idx & (INNER - 1);
    const int bt = idx >> 11;                          // INNER = 2^11
    const int t  = bt & (SEQ - 1);
    const int b  = bt >> 11;
    float acc = cb[d];
    #pragma unroll
    for (int j = 0; j < 4; ++j) {
        const int tt = t - 3 + j;
        if (tt >= 0)
            acc += ck[j * INNER + d] * uz[((size_t)(b * SEQ + tt)) * (2 * INNER) + d];
    }
    u[idx] = siluf_(acc);
}

// ================= selective scan (+ gate, in-place over u) =================
__global__ void scan_kernel(const float* __restrict__ dt,
                            const float* __restrict__ uin,
                            const float* __restrict__ bmat,
                            const float* __restrict__ cmat,
                            const float* __restrict__ uz,
                            const float* __restrict__ a_log,
                            const float* __restrict__ dvec,
                            float* __restrict__ yout) {   // yout aliases uin (safe: RAW then WAW per element, same thread)
    const int ch = blockIdx.x * 256 + threadIdx.x;        // 0..4095
    const int b = ch >> 11;
    const int d = ch & (INNER - 1);
    float a[DSTATE], h[DSTATE];
    #pragma unroll
    for (int s = 0; s < DSTATE; ++s) {
        a[s] = -__expf(a_log[d * DSTATE + s]);
        h[s] = 0.f;
    }
    const float Dd = dvec[d];
    for (int t = 0; t < SEQ; ++t) {
        const int row = b * SEQ + t;
        const float dtv = dt[(size_t)row * INNER + d];
        const float uv  = uin[(size_t)row * INNER + d];
        float y = 0.f;
        #pragma unroll
        for (int s = 0; s < DSTATE; ++s) {
            const float da = __expf(dtv * a[s]);
            h[s] = da * h[s] + dtv * uv * bmat[row * DSTATE + s];
            y += h[s] * cmat[row * DSTATE + s];
        }
        y += uv * Dd;
        const float zv = uz[(size_t)row * (2 * INNER) + INNER + d];
        yout[(size_t)row * INNER + d] = y * siluf_(zv);
    }
}

// ================= launcher =================
extern "C" void kernel_launch(void* const* d_in, const int* in_sizes, int n_in,
                              void* d_out, int out_size, void* d_ws, size_t ws_size,
                              hipStream_t stream) {
    const float* x          = (const float*)d_in[0];
    const float* norm_scale = (const float*)d_in[1];
    const float* in_proj_w  = (const float*)d_in[2];
    const float* in_proj_b  = (const float*)d_in[3];
    const float* out_proj_w = (const float*)d_in[4];
    const float* out_proj_b = (const float*)d_in[5];
    const float* dt_in_w    = (const float*)d_in[6];
    const float* dt_in_b    = (const float*)d_in[7];
    const float* dt_out_w   = (const float*)d_in[8];
    const float* dt_out_b   = (const float*)d_in[9];
    const float* b_proj_w   = (const float*)d_in[10];
    const float* b_proj_b   = (const float*)d_in[11];
    const float* c_proj_w   = (const float*)d_in[12];
    const float* c_proj_b   = (const float*)d_in[13];
    const float* conv_k     = (const float*)d_in[14];
    const float* conv_b     = (const float*)d_in[15];
    const float* a_log      = (const float*)d_in[16];
    const float* dvec       = (const float*)d_in[17];
    const float* dt_bias    = (const float*)d_in[18];

    float* ws   = (float*)d_ws;
    float* xn   = ws + XN_OFF;
    float* uz   = ws + UZ_OFF;
    float* u    = ws + U_OFF;
    float* dtr  = ws + DTR_OFF;
    float* dt   = ws + DT_OFF;
    float* bmat = ws + BM_OFF;
    float* cmat = ws + CM_OFF;
    float* out  = (float*)d_out;

    // 1. RMSNorm
    rmsnorm_kernel<<<MROWS, 256, 0, stream>>>(x, norm_scale, xn);

    // 2. in_proj: [4096,1024] @ [1024,4096] -> uz   (wave tile 16x128)
    gemm_wmma_f32<8, 0><<<dim3(2 * INNER / 128, MROWS / 64), 128, 0, stream>>>(
        xn, in_proj_w, in_proj_b, nullptr, nullptr, uz, MROWS, 2 * INNER, DMODEL);

    // 3. causal conv(4) + SiLU -> u
    conv_silu_kernel<<<(MROWS * INNER) / 256, 256, 0, stream>>>(uz, conv_k, conv_b, u);

    // 4a. dt rank-reduce: [4096,2048] @ [2048,64] -> dtr
    gemm_wmma_f32<1, 0><<<dim3(DTRANK / 16, MROWS / 64), 128, 0, stream>>>(
        u, dt_in_w, dt_in_b, nullptr, nullptr, dtr, MROWS, DTRANK, INNER);

    // 4b. dt expand + softplus/clip: [4096,64] @ [64,2048] -> dt
    gemm_wmma_f32<8, 2><<<dim3(INNER / 128, MROWS / 64), 128, 0, stream>>>(
        dtr, dt_out_w, dt_out_b, dt_bias, nullptr, dt, MROWS, INNER, DTRANK);

    // 5. B / C projections with tanh: [4096,2048] @ [2048,16]
    gemm_wmma_f32<1, 1><<<dim3(1, MROWS / 64), 128, 0, stream>>>(
        u, b_proj_w, b_proj_b, nullptr, nullptr, bmat, MROWS, DSTATE, INNER);
    gemm_wmma_f32<1, 1><<<dim3(1, MROWS / 64), 128, 0, stream>>>(
        u, c_proj_w, c_proj_b, nullptr, nullptr, cmat, MROWS, DSTATE, INNER);

    // 6. selective scan + gate (overwrites u with y*silu(z))
    scan_kernel<<<(BATCH * INNER) / 256, 256, 0, stream>>>(
        dt, u, bmat, cmat, uz, a_log, dvec, u);

    // 7. out_proj + bias + residual: [4096,2048] @ [2048,1024] -> out
    gemm_wmma_f32<8, 0><<<dim3(DMODEL / 128, MROWS / 64), 128, 0, stream>>>(
        u, out_proj_w, out_proj_b, nullptr, x, out, MROWS, DMODEL, INNER);
}